// TopkSelectorLayer_16965120819288
// MI455X (gfx1250) — compile-verified
//
#include <hip/hip_runtime.h>
#include <stdint.h>

// ---------------------------------------------------------------------------
// Masked row-wise top-k (B=128, S=262144, K=2048) via 2-pass radix select.
//
// Roofline: mandatory traffic = 134MB scan for the histogram pass. The whole
// score matrix (134MB) fits in the 192MB L2, so the second (gather) pass hits
// L2. To saturate 23.3 TB/s HBM on the first pass we split each row across
// SPLIT=4 workgroups (512 streaming WGs, 32 wave32 waves each), all fed by
// CDNA5 async global->LDS loads (ASYNCcnt-tracked, double buffered).
//
// Pipeline (stream-ordered kernels, graph-capture safe):
//   K1: per-(row,segment) LDS histogram (top 13 key bits) -> private ws slice
//   K2: per-row slice reduction + block scan -> threshold key, reset counters
//   K3: per-(row,segment) async-streamed gather; LDS aggregation; ONE global
//       atomic reservation per WG (4 atomics/row) -> global candidate list
//   K4: per-row bitonic sort (descending, tie -> lower index) -> top-K indices
//
// Fallback: if ws_size is too small, use the self-contained one-WG-per-row
// kernel (identical algorithm, fused).
// ---------------------------------------------------------------------------

#define BATCH     128
#define SEQ_LEN   262144
#define TOPK      2048
#define NTHREADS  1024          // 32 wave32 waves
#define TILE      8192          // floats per streamed tile (32 KB)
#define TILE_BYTES (TILE * 4)
#define NTILES    (SEQ_LEN / TILE)
#define NBINS     8192          // top-13-bit histogram
#define CAND_CAP  8192
#define SPLIT     4
#define SEG       (SEQ_LEN / SPLIT)   // 65536 floats per segment
#define SEG_TILES (SEG / TILE)        // 8 tiles per segment

// ---- workspace layout (bytes) ----
#define WS_HIST_OFF  0
#define WS_HIST_SZ   ((size_t)BATCH * SPLIT * NBINS * 4)        // 16 MB
#define WS_CAND_OFF  (WS_HIST_OFF + WS_HIST_SZ)
#define WS_CAND_SZ   ((size_t)BATCH * CAND_CAP * 8)             // 8 MB
#define WS_CNT_OFF   (WS_CAND_OFF + WS_CAND_SZ)
#define WS_CNT_SZ    ((size_t)512)
#define WS_THR_OFF   (WS_CNT_OFF + WS_CNT_SZ)
#define WS_THR_SZ    ((size_t)512)
#define WS_NEED      (WS_THR_OFF + WS_THR_SZ)

// ---- fallback-kernel LDS layout ----
#define OFF_STAGE 0
#define OFF_UNION (2 * TILE_BYTES)
#define OFF_SCAN  (OFF_UNION + CAND_CAP * 8)
#define OFF_CTRL  (OFF_SCAN + NTHREADS * 4)
#define SMEM_FALLBACK (OFF_CTRL + 64)

// ---- split-kernel LDS sizes ----
#define SMEM_K1 (2 * TILE_BYTES + NBINS * 4)             // 96 KB
#define SMEM_K3 (2 * TILE_BYTES + CAND_CAP * 8 + 64)     // 128 KB + ctrl
#define SMEM_K4 (CAND_CAP * 8)                           // 64 KB

extern __shared__ unsigned char smem_raw[];

// Issue one tile's async global->LDS loads: each thread moves 2 x 16B.
// Instruction 0 covers bytes [0,16384) of the tile, instruction 1 the rest,
// so each wave32 issue is a fully contiguous 512B burst.
__device__ __forceinline__ void issue_tile(const float* __restrict__ segPtr,
                                           int tile, int buf,
                                           uint32_t ldsBase, int tid) {
    uint32_t l0 = ldsBase + (uint32_t)(OFF_STAGE + buf * TILE_BYTES + tid * 16);
    const char* g0 = (const char*)(segPtr + (size_t)tile * TILE) + tid * 16;
    asm volatile("global_load_async_to_lds_b128 %0, %1, off"
                 :: "v"(l0), "v"((uint64_t)(uintptr_t)g0) : "memory");
    asm volatile("global_load_async_to_lds_b128 %0, %1, off"
                 :: "v"(l0 + (uint32_t)(TILE_BYTES / 2)),
                    "v"((uint64_t)(uintptr_t)(g0 + TILE_BYTES / 2)) : "memory");
}

__device__ __forceinline__ void wait_async_2() {
    asm volatile("s_wait_asynccnt 0x2" ::: "memory");   // previous tile landed
}
__device__ __forceinline__ void wait_async_0() {
    asm volatile("s_wait_asynccnt 0x0" ::: "memory");
}

// Order-preserving float -> u32 key (total order matches float compare).
__device__ __forceinline__ uint32_t monokey(float f) {
    uint32_t b = __float_as_uint(f);
    return b ^ ((uint32_t)((int32_t)b >> 31) | 0x80000000u);
}

// ===========================================================================
// K1: per-(row,segment) histogram of the top 13 key bits.
// ===========================================================================
__global__ __launch_bounds__(NTHREADS)
void k1_hist(const float* __restrict__ scores,
             const int*  __restrict__ starts,
             const int*  __restrict__ ends,
             uint32_t*   __restrict__ ghist) {
    const int split = (int)blockIdx.x;
    const int row   = (int)blockIdx.y;
    const int tid   = (int)threadIdx.x;
    const uint32_t ldsBase = (uint32_t)__builtin_amdgcn_groupstaticsize();

    float*    stage = (float*)(smem_raw + OFF_STAGE);
    uint32_t* hist  = (uint32_t*)(smem_raw + 2 * TILE_BYTES);

    const float* segPtr = scores + (size_t)row * SEQ_LEN + (size_t)split * SEG;
    const int segBase = split * SEG;
    const int start = starts[row];
    const int end   = ends[row];

    for (int i = tid; i < NBINS; i += NTHREADS) hist[i] = 0u;
    __syncthreads();

    issue_tile(segPtr, 0, 0, ldsBase, tid);
    for (int t = 0; t < SEG_TILES; ++t) {
        if (t + 1 < SEG_TILES) {
            issue_tile(segPtr, t + 1, (t + 1) & 1, ldsBase, tid);
            wait_async_2();
        } else {
            wait_async_0();
        }
        __syncthreads();

        const float* buf = stage + (size_t)(t & 1) * TILE;
#pragma unroll
        for (int half = 0; half < 2; ++half) {
            const int lbase = half * (TILE / 2) + tid * 4;
            const int gbase = segBase + t * TILE + lbase;
            float4 v = *(const float4*)(buf + lbase);
            float vals[4] = {v.x, v.y, v.z, v.w};
#pragma unroll
            for (int j = 0; j < 4; ++j) {
                const int p = gbase + j;
                if (p >= start && p < end)
                    atomicAdd(&hist[monokey(vals[j]) >> 19], 1u);
            }
        }
        __syncthreads();
    }

    // plain stores to this WG's private slice (no global atomics)
    uint32_t* dst = ghist + ((size_t)row * SPLIT + split) * NBINS;
    for (int i = tid; i < NBINS; i += NTHREADS) dst[i] = hist[i];
}

// ===========================================================================
// K2: per-row slice reduction + suffix-count crossing -> threshold key.
// ===========================================================================
__global__ __launch_bounds__(NTHREADS)
void k2_thresh(const uint32_t* __restrict__ ghist,
               uint32_t*       __restrict__ thrArr,
               uint32_t*       __restrict__ candCnt) {
    __shared__ uint32_t scanBuf[NTHREADS];
    __shared__ uint32_t thrBin;
    const int row = (int)blockIdx.x;
    const int tid = (int)threadIdx.x;
    const uint32_t* h = ghist + (size_t)row * SPLIT * NBINS;

    uint32_t cnt8[8];
    uint32_t s = 0;
#pragma unroll
    for (int j = 0; j < 8; ++j) {
        const int r = tid * 8 + j;            // r-order: descending bins
        const int bin = NBINS - 1 - r;
        uint32_t c = 0;
#pragma unroll
        for (int sl = 0; sl < SPLIT; ++sl) c += h[(size_t)sl * NBINS + bin];
        cnt8[j] = c;
        s += c;
    }
    scanBuf[tid] = s;
    __syncthreads();
    for (int off = 1; off < NTHREADS; off <<= 1) {
        const uint32_t y = (tid >= off) ? scanBuf[tid - off] : 0u;
        __syncthreads();
        scanBuf[tid] += y;
        __syncthreads();
    }
    uint32_t run = (tid > 0) ? scanBuf[tid - 1] : 0u;
#pragma unroll
    for (int j = 0; j < 8; ++j) {
        const uint32_t nxt = run + cnt8[j];
        if (run < (uint32_t)TOPK && nxt >= (uint32_t)TOPK)
            thrBin = (uint32_t)(NBINS - 1 - (tid * 8 + j));    // unique writer
        run = nxt;
    }
    __syncthreads();
    if (tid == 0) {
        thrArr[row]  = thrBin << 19;    // candidate iff key >= thrKey
        candCnt[row] = 0u;              // reset reservation counter
    }
}

// ===========================================================================
// K3: per-(row,segment) gather: async stream + LDS aggregation + one global
//     atomic reservation per workgroup.
// ===========================================================================
__global__ __launch_bounds__(NTHREADS)
void k3_gather(const float*    __restrict__ scores,
               const int*      __restrict__ starts,
               const int*      __restrict__ ends,
               const uint32_t* __restrict__ thrArr,
               uint32_t*       __restrict__ candCnt,
               uint64_t*       __restrict__ gcand) {
    const int split = (int)blockIdx.x;
    const int row   = (int)blockIdx.y;
    const int tid   = (int)threadIdx.x;
    const uint32_t ldsBase = (uint32_t)__builtin_amdgcn_groupstaticsize();

    float*    stage = (float*)(smem_raw + OFF_STAGE);
    uint64_t* lcand = (uint64_t*)(smem_raw + 2 * TILE_BYTES);
    uint32_t* ctrl  = (uint32_t*)(smem_raw + 2 * TILE_BYTES + CAND_CAP * 8);

    const float* segPtr = scores + (size_t)row * SEQ_LEN + (size_t)split * SEG;
    const int segBase = split * SEG;
    const int start = starts[row];
    const int end   = ends[row];
    const uint32_t thrKey = thrArr[row];

    if (tid == 0) ctrl[0] = 0u;
    __syncthreads();

    issue_tile(segPtr, 0, 0, ldsBase, tid);
    for (int t = 0; t < SEG_TILES; ++t) {
        if (t + 1 < SEG_TILES) {
            issue_tile(segPtr, t + 1, (t + 1) & 1, ldsBase, tid);
            wait_async_2();
        } else {
            wait_async_0();
        }
        __syncthreads();

        const float* buf = stage + (size_t)(t & 1) * TILE;
#pragma unroll
        for (int half = 0; half < 2; ++half) {
            const int lbase = half * (TILE / 2) + tid * 4;
            const int gbase = segBase + t * TILE + lbase;
            float4 v = *(const float4*)(buf + lbase);
            float vals[4] = {v.x, v.y, v.z, v.w};
#pragma unroll
            for (int j = 0; j < 4; ++j) {
                const int p = gbase + j;
                if (p >= start && p < end) {
                    const uint32_t key = monokey(vals[j]);
                    if (key >= thrKey) {
                        const uint32_t pos = atomicAdd(&ctrl[0], 1u);
                        if (pos < CAND_CAP)
                            // pack: key desc; equal keys -> lower idx first
                            lcand[pos] = ((uint64_t)key << 18) |
                                         (uint64_t)((SEQ_LEN - 1) - p);
                    }
                }
            }
        }
        __syncthreads();
    }

    uint32_t n = ctrl[0];
    if (n > CAND_CAP) n = CAND_CAP;
    if (tid == 0) ctrl[1] = atomicAdd(&candCnt[row], n);   // one atomic per WG
    __syncthreads();
    const uint32_t base = ctrl[1];
    uint64_t* dst = gcand + (size_t)row * CAND_CAP;
    for (uint32_t i = (uint32_t)tid; i < n; i += NTHREADS) {
        const uint32_t g = base + i;
        if (g < CAND_CAP) dst[g] = lcand[i];
    }
}

// ===========================================================================
// K4: per-row pad + bitonic sort (descending) + emit top-K indices.
// ===========================================================================
__global__ __launch_bounds__(NTHREADS)
void k4_sort(const uint32_t* __restrict__ candCnt,
             const uint64_t* __restrict__ gcand,
             int*            __restrict__ out) {
    const int row = (int)blockIdx.x;
    const int tid = (int)threadIdx.x;
    uint64_t* cand = (uint64_t*)smem_raw;

    uint32_t total = candCnt[row];
    if (total > CAND_CAP) total = CAND_CAP;
    const uint64_t* src = gcand + (size_t)row * CAND_CAP;
    for (int i = tid; i < CAND_CAP; i += NTHREADS)
        cand[i] = ((uint32_t)i < total) ? src[i] : 0ull;
    __syncthreads();

    for (unsigned k = 2; k <= CAND_CAP; k <<= 1) {
        for (unsigned j = k >> 1; j > 0; j >>= 1) {
            for (unsigned i = (unsigned)tid; i < CAND_CAP; i += NTHREADS) {
                const unsigned ixj = i ^ j;
                if (ixj > i) {
                    const uint64_t a = cand[i];
                    const uint64_t b = cand[ixj];
                    const bool upper = ((i & k) == 0);
                    if (upper ? (a < b) : (a > b)) {
                        cand[i] = b;
                        cand[ixj] = a;
                    }
                }
            }
            __syncthreads();
        }
    }

    for (int i = tid; i < TOPK; i += NTHREADS) {
        const uint64_t pk = cand[i];
        out[(size_t)row * TOPK + i] =
            (int)((SEQ_LEN - 1) - (uint32_t)(pk & 0x3FFFFu));
    }
}

// ===========================================================================
// Fallback: fused one-workgroup-per-row version (no workspace needed).
// ===========================================================================
__global__ __launch_bounds__(NTHREADS)
void topk_fused_kernel(const float* __restrict__ scores,
                       const int*  __restrict__ starts,
                       const int*  __restrict__ ends,
                       int*        __restrict__ out) {
    const int row = (int)blockIdx.x;
    const int tid = (int)threadIdx.x;
    const uint32_t ldsBase = (uint32_t)__builtin_amdgcn_groupstaticsize();

    float*     stage   = (float*)(smem_raw + OFF_STAGE);
    uint32_t*  hist    = (uint32_t*)(smem_raw + OFF_UNION);
    uint64_t*  cand    = (uint64_t*)(smem_raw + OFF_UNION);   // union with hist
    uint32_t*  scanBuf = (uint32_t*)(smem_raw + OFF_SCAN);
    uint32_t*  ctrl    = (uint32_t*)(smem_raw + OFF_CTRL);

    const float* rowPtr = scores + (size_t)row * SEQ_LEN;
    const int start = starts[row];
    const int end   = ends[row];

    for (int i = tid; i < NBINS; i += NTHREADS) hist[i] = 0u;
    __syncthreads();

    issue_tile(rowPtr, 0, 0, ldsBase, tid);
    for (int t = 0; t < NTILES; ++t) {
        if (t + 1 < NTILES) {
            issue_tile(rowPtr, t + 1, (t + 1) & 1, ldsBase, tid);
            wait_async_2();
        } else {
            wait_async_0();
        }
        __syncthreads();
        const float* buf = stage + (size_t)(t & 1) * TILE;
#pragma unroll
        for (int half = 0; half < 2; ++half) {
            const int lbase = half * (TILE / 2) + tid * 4;
            const int gbase = t * TILE + lbase;
            float4 v = *(const float4*)(buf + lbase);
            float vals[4] = {v.x, v.y, v.z, v.w};
#pragma unroll
            for (int j = 0; j < 4; ++j) {
                const int p = gbase + j;
                if (p >= start && p < end)
                    atomicAdd(&hist[monokey(vals[j]) >> 19], 1u);
            }
        }
        __syncthreads();
    }

    uint32_t cnt8[8];
    uint32_t s = 0;
#pragma unroll
    for (int j = 0; j < 8; ++j) {
        const int r = tid * 8 + j;
        const uint32_t c = hist[NBINS - 1 - r];
        cnt8[j] = c;
        s += c;
    }
    scanBuf[tid] = s;
    __syncthreads();
    for (int off = 1; off < NTHREADS; off <<= 1) {
        const uint32_t y = (tid >= off) ? scanBuf[tid - off] : 0u;
        __syncthreads();
        scanBuf[tid] += y;
        __syncthreads();
    }
    uint32_t run = (tid > 0) ? scanBuf[tid - 1] : 0u;
#pragma unroll
    for (int j = 0; j < 8; ++j) {
        const uint32_t nxt = run + cnt8[j];
        if (run < (uint32_t)TOPK && nxt >= (uint32_t)TOPK)
            ctrl[0] = (uint32_t)(NBINS - 1 - (tid * 8 + j));
        run = nxt;
    }
    if (tid == 0) ctrl[1] = 0u;
    __syncthreads();
    const uint32_t thrKey = ctrl[0] << 19;

    issue_tile(rowPtr, 0, 0, ldsBase, tid);
    for (int t = 0; t < NTILES; ++t) {
        if (t + 1 < NTILES) {
            issue_tile(rowPtr, t + 1, (t + 1) & 1, ldsBase, tid);
            wait_async_2();
        } else {
            wait_async_0();
        }
        __syncthreads();
        const float* buf = stage + (size_t)(t & 1) * TILE;
#pragma unroll
        for (int half = 0; half < 2; ++half) {
            const int lbase = half * (TILE / 2) + tid * 4;
            const int gbase = t * TILE + lbase;
            float4 v = *(const float4*)(buf + lbase);
            float vals[4] = {v.x, v.y, v.z, v.w};
#pragma unroll
            for (int j = 0; j < 4; ++j) {
                const int p = gbase + j;
                if (p >= start && p < end) {
                    const uint32_t key = monokey(vals[j]);
                    if (key >= thrKey) {
                        const uint32_t pos = atomicAdd(&ctrl[1], 1u);
                        if (pos < CAND_CAP)
                            cand[pos] = ((uint64_t)key << 18) |
                                        (uint64_t)((SEQ_LEN - 1) - p);
                    }
                }
            }
        }
        __syncthreads();
    }

    uint32_t total = ctrl[1];
    if (total > CAND_CAP) total = CAND_CAP;
    for (int i = tid; i < CAND_CAP; i += NTHREADS)
        if ((uint32_t)i >= total) cand[i] = 0ull;
    __syncthreads();

    for (unsigned k = 2; k <= CAND_CAP; k <<= 1) {
        for (unsigned j = k >> 1; j > 0; j >>= 1) {
            for (unsigned i = (unsigned)tid; i < CAND_CAP; i += NTHREADS) {
                const unsigned ixj = i ^ j;
                if (ixj > i) {
                    const uint64_t a = cand[i];
                    const uint64_t b = cand[ixj];
                    const bool upper = ((i & k) == 0);
                    if (upper ? (a < b) : (a > b)) {
                        cand[i] = b;
                        cand[ixj] = a;
                    }
                }
            }
            __syncthreads();
        }
    }

    for (int i = tid; i < TOPK; i += NTHREADS) {
        const uint64_t pk = cand[i];
        out[(size_t)row * TOPK + i] =
            (int)((SEQ_LEN - 1) - (uint32_t)(pk & 0x3FFFFu));
    }
}

// ===========================================================================
extern "C" void kernel_launch(void* const* d_in, const int* in_sizes, int n_in,
                              void* d_out, int out_size, void* d_ws, size_t ws_size,
                              hipStream_t stream) {
    (void)in_sizes; (void)n_in; (void)out_size;
    const float* scores = (const float*)d_in[0];
    const int*   starts = (const int*)d_in[1];
    const int*   ends   = (const int*)d_in[2];
    int*         out    = (int*)d_out;

    if (d_ws != nullptr && ws_size >= WS_NEED) {
        uint32_t* ghist   = (uint32_t*)((char*)d_ws + WS_HIST_OFF);
        uint64_t* gcand   = (uint64_t*)((char*)d_ws + WS_CAND_OFF);
        uint32_t* candCnt = (uint32_t*)((char*)d_ws + WS_CNT_OFF);
        uint32_t* thrArr  = (uint32_t*)((char*)d_ws + WS_THR_OFF);

        dim3 gridScan(SPLIT, BATCH);
        k1_hist  <<<gridScan,   dim3(NTHREADS), SMEM_K1, stream>>>(
            scores, starts, ends, ghist);
        k2_thresh<<<dim3(BATCH), dim3(NTHREADS), 0,       stream>>>(
            ghist, thrArr, candCnt);
        k3_gather<<<gridScan,   dim3(NTHREADS), SMEM_K3, stream>>>(
            scores, starts, ends, thrArr, candCnt, gcand);
        k4_sort  <<<dim3(BATCH), dim3(NTHREADS), SMEM_K4, stream>>>(
            candCnt, gcand, out);
    } else {
        topk_fused_kernel<<<dim3(BATCH), dim3(NTHREADS), SMEM_FALLBACK, stream>>>(
            scores, starts, ends, out);
    }
}